// TaskAlignedAssigner_7507602833590
// MI455X (gfx1250) — compile-verified
//
#include <hip/hip_runtime.h>
#include <math.h>

#define AN 8400
#define BN 64
#define GN 32
#define CN 80
#define CH 84
#define K_TOP 13

typedef float v2f __attribute__((ext_vector_type(2)));
typedef float v8f __attribute__((ext_vector_type(8)));
typedef __attribute__((address_space(3))) char lds_char_t;

__device__ __forceinline__ unsigned lds_off(void* p) {
  return (unsigned)(unsigned long long)(lds_char_t*)p;
}
// CDNA5 async global->LDS copy (16B per lane), tracked by ASYNCcnt
__device__ __forceinline__ void async_copy_b128(void* lds_dst, const void* gsrc) {
  asm volatile("global_load_async_to_lds_b128 %0, %1, off"
               :: "v"(lds_off(lds_dst)), "v"((unsigned long long)(size_t)gsrc)
               : "memory");
}
__device__ __forceinline__ void wait_async0() {
  asm volatile("s_wait_asynccnt 0" ::: "memory");
}

// ---------------------------------------------------------------------------
// Kernel 1: per (b, 16-anchor tile): softmax, WMMA one-hot class gather,
// IoU, align metric, best-iou gt, inside mask. One wave per workgroup.
// ---------------------------------------------------------------------------
__global__ __launch_bounds__(32) void k1_score_align(
    const float* __restrict__ preds,      // (B, 84, A)
    const float* __restrict__ gtb,        // (B, G, 4)
    const int*   __restrict__ gtc,        // (B, G)
    const float* __restrict__ anc,        // (A, 2)
    float*    __restrict__ align_ws,      // (B, G, A)
    unsigned* __restrict__ topk_ws,       // (B, A) zeroed here
    unsigned* __restrict__ inside_ws,     // (B, A)
    int*      __restrict__ bestg_ws,      // (B, A)
    float*    __restrict__ out_sum)       // scalar: zeroed here
{
  __shared__ float gtb_lds[GN * 4];
  __shared__ float pb_lds[16 * 4];
  __shared__ float denom_lds[16];
  __shared__ float iou_lds[16 * 33];     // pitch 33: bank-conflict-free
  __shared__ float al_lds[16 * 33];

  const int L  = threadIdx.x;
  const int hh = L >> 4;                 // half-wave id (0/1)
  const int la = L & 15;
  const int b  = blockIdx.y;
  const int a0 = blockIdx.x * 16;
  const int a  = a0 + la;

  if (blockIdx.x == 0 && b == 0 && L == 0) out_sum[0] = 0.0f;

  // async copy this batch's GT boxes into LDS (32 lanes x 16B = 512B)
  async_copy_b128(&gtb_lds[L * 4], gtb + (size_t)b * GN * 4 + L * 4);

  const float* P = preds + (size_t)b * CH * AN;

  // box decode (duplicated across halves; coalesced 64B segments)
  const float ax = anc[2 * a], ay = anc[2 * a + 1];
  const float dl = P[0 * AN + a], dt = P[1 * AN + a];
  const float dr = P[2 * AN + a], db = P[3 * AN + a];
  const float px1 = ax - dl, py1 = ay - dt, px2 = ax + dr, py2 = ay + db;

  const int gc0 = gtc[b * GN + la];
  const int gc1 = gtc[b * GN + 16 + la];

  // Load logits directly in WMMA A-fragment layout:
  // lane holds classes {4j+2*hh, 4j+2*hh+1} of anchor a0+la, j=0..19.
  float ea[20], eb[20];
#pragma unroll
  for (int j = 0; j < 20; ++j) {
    const int c = 4 * j + 2 * hh;
    ea[j] = P[(size_t)(4 + c) * AN + a];
    eb[j] = P[(size_t)(5 + c) * AN + a];
  }
  float mx = -3.4e38f;
#pragma unroll
  for (int j = 0; j < 20; ++j) mx = fmaxf(mx, fmaxf(ea[j], eb[j]));
  mx = fmaxf(mx, __shfl_xor(mx, 16, 32));   // combine halves -> full 80-class max
  float s = 0.f;
#pragma unroll
  for (int j = 0; j < 20; ++j) {
    ea[j] = __expf(ea[j] - mx);
    eb[j] = __expf(eb[j] - mx);
    s += ea[j] + eb[j];
  }
  s += __shfl_xor(s, 16, 32);               // full softmax denominator

  if (hh == 0) {
    denom_lds[la] = s;
    pb_lds[la * 4 + 0] = px1; pb_lds[la * 4 + 1] = py1;
    pb_lds[la * 4 + 2] = px2; pb_lds[la * 4 + 3] = py2;
  }
  wait_async0();
  __syncthreads();

  // One-hot gather as matmul: exp(16x80) . onehot(80x32) via 40 WMMAs (K=4).
  v8f acc0 = {0.f, 0.f, 0.f, 0.f, 0.f, 0.f, 0.f, 0.f};
  v8f acc1 = {0.f, 0.f, 0.f, 0.f, 0.f, 0.f, 0.f, 0.f};
  const int kb = 2 * hh;
#pragma unroll
  for (int j = 0; j < 20; ++j) {
    const int c0 = 4 * j + kb;
    v2f af = {ea[j], eb[j]};
    v2f b0 = {gc0 == c0 ? 1.f : 0.f, gc0 == c0 + 1 ? 1.f : 0.f};
    v2f b1 = {gc1 == c0 ? 1.f : 0.f, gc1 == c0 + 1 ? 1.f : 0.f};
    acc0 = __builtin_amdgcn_wmma_f32_16x16x4_f32(false, af, false, b0,
                                                 (short)0, acc0, false, false);
    acc1 = __builtin_amdgcn_wmma_f32_16x16x4_f32(false, af, false, b1,
                                                 (short)0, acc1, false, false);
  }

  // D layout: vgpr r, lane L -> anchor m=r+8*hh, gt n=la (tile0) / 16+la (tile1)
#pragma unroll
  for (int r = 0; r < 8; ++r) {
    const int m = r + 8 * hh;
    const float den = denom_lds[m];
    const float qx1 = pb_lds[m * 4 + 0], qy1 = pb_lds[m * 4 + 1];
    const float qx2 = pb_lds[m * 4 + 2], qy2 = pb_lds[m * 4 + 3];
    const float areaA = (qx2 - qx1) * (qy2 - qy1);
#pragma unroll
    for (int t = 0; t < 2; ++t) {
      const int n = 16 * t + la;
      const float sc = (t == 0 ? acc0[r] : acc1[r]) / den;
      const float gx1 = gtb_lds[n * 4 + 0], gy1 = gtb_lds[n * 4 + 1];
      const float gx2 = gtb_lds[n * 4 + 2], gy2 = gtb_lds[n * 4 + 3];
      const float iw = fmaxf(fminf(qx2, gx2) - fmaxf(qx1, gx1), 0.f);
      const float ih = fmaxf(fminf(qy2, gy2) - fmaxf(qy1, gy1), 0.f);
      const float inter = iw * ih;
      const float areaB = (gx2 - gx1) * (gy2 - gy1);
      const float iou = inter / (areaA + areaB - inter + 1e-7f);
      const float al = sqrtf(fmaxf(iou, 1e-12f)) * sqrtf(fmaxf(sc, 1e-12f));
      iou_lds[m * 33 + n] = iou;
      al_lds[m * 33 + n] = al;
    }
  }
  __syncthreads();

  // Per-anchor reductions: best-iou gt (ties -> lowest g) + inside-box mask.
  float bestv = -1.f;
  int bg = 0;
  unsigned ins = 0;
  const int gbase = 16 * hh;
#pragma unroll
  for (int gg = 0; gg < 16; ++gg) {
    const int g = gbase + gg;
    const float v = iou_lds[la * 33 + g];
    if (v > bestv) { bestv = v; bg = g; }
    const float gx1 = gtb_lds[g * 4 + 0], gy1 = gtb_lds[g * 4 + 1];
    const float gx2 = gtb_lds[g * 4 + 2], gy2 = gtb_lds[g * 4 + 3];
    const float dmin = fminf(fminf(ax - gx1, ay - gy1),
                             fminf(gx2 - ax, gy2 - ay));
    if (dmin > 1e-9f) ins |= (1u << g);
  }
  const float ov = __shfl_xor(bestv, 16, 32);
  const int   og = __shfl_xor(bg, 16, 32);
  ins |= __shfl_xor(ins, 16, 32);
  if (ov > bestv || (ov == bestv && og < bg)) { bestv = ov; bg = og; }

  if (hh == 0) {
    const size_t o = (size_t)b * AN + a;
    bestg_ws[o] = bg;
    inside_ws[o] = ins;
    topk_ws[o] = 0u;   // zero for kernel 2's atomicOr
  }

  // store align in (B, G, A) layout (g-major, anchors contiguous)
#pragma unroll
  for (int j = 0; j < 16; ++j) {
    const int g = 2 * j + hh;
    align_ws[((size_t)b * GN + g) * AN + a0 + la] = al_lds[la * 33 + g];
  }
}

// ---------------------------------------------------------------------------
// Kernel 2: top-13 anchors per (b,g); scatter bits with atomicOr.
// ---------------------------------------------------------------------------
__global__ __launch_bounds__(256) void k2_topk(
    const float* __restrict__ align_ws, unsigned* __restrict__ topk_ws)
{
  __shared__ float row[AN];          // 33.6 KB: full align row resident in LDS
  __shared__ float rv[256];
  __shared__ int   ri[256];
  __shared__ int   sel[K_TOP];

  const int tid = threadIdx.x;
  const int g = blockIdx.x, b = blockIdx.y;
  const float* src = align_ws + ((size_t)b * GN + g) * AN;

  for (int i = tid; i < AN / 4; i += 256)      // async bulk copy, 16B/lane
    async_copy_b128(&row[i * 4], src + i * 4);
  wait_async0();
  __syncthreads();

  for (int k = 0; k < K_TOP; ++k) {
    float bv = -2.f; int bi = AN;
    for (int i = tid; i < AN; i += 256) {
      const float v = row[i];
      if (v > bv) { bv = v; bi = i; }          // ascending scan -> lowest idx
    }
    rv[tid] = bv; ri[tid] = bi;
    __syncthreads();
    for (int s2 = 128; s2 > 0; s2 >>= 1) {
      if (tid < s2) {
        const float v2 = rv[tid + s2]; const int i2 = ri[tid + s2];
        if (v2 > rv[tid] || (v2 == rv[tid] && i2 < ri[tid])) {
          rv[tid] = v2; ri[tid] = i2;
        }
      }
      __syncthreads();
    }
    if (tid == 0) { const int w = ri[0]; sel[k] = w; row[w] = -1.f; }
    __syncthreads();
  }
  if (tid < K_TOP)
    atomicOr(&topk_ws[(size_t)b * AN + sel[tid]], 1u << g);
}

// ---------------------------------------------------------------------------
// Kernel 3: resolve masks, write tb/tc/ts/fg and fg-sum.
// ---------------------------------------------------------------------------
__global__ __launch_bounds__(256) void k3_finalize(
    const float* __restrict__ gtb, const int* __restrict__ gtc,
    const float* __restrict__ align_ws, const unsigned* __restrict__ topk_ws,
    const unsigned* __restrict__ inside_ws, const int* __restrict__ bestg_ws,
    float* __restrict__ out)
{
  const size_t N_TB = (size_t)BN * AN * 4;
  const size_t N_BA = (size_t)BN * AN;

  __shared__ int cnt_lds;
  if (threadIdx.x == 0) cnt_lds = 0;
  __syncthreads();

  const int b = blockIdx.y;
  const int a = blockIdx.x * 256 + threadIdx.x;
  if (a < AN) {
    const size_t o = (size_t)b * AN + a;
    unsigned m = topk_ws[o] & inside_ws[o];
    if (__popc(m) > 1) m = 1u << (unsigned)bestg_ws[o];   // multi -> best iou
    const int fg = (m != 0u) ? 1 : 0;
    const int tgt = fg ? (__ffs(m) - 1) : 0;              // first set bit
    const float f = (float)fg;

    const float* g4 = gtb + ((size_t)b * GN + tgt) * 4;
    float4 tb;
    tb.x = g4[0] * f; tb.y = g4[1] * f; tb.z = g4[2] * f; tb.w = g4[3] * f;
    reinterpret_cast<float4*>(out)[o] = tb;
    out[N_TB + o]            = fg ? (float)gtc[b * GN + tgt] : 0.f;
    out[N_TB + N_BA + o]     = align_ws[((size_t)b * GN + tgt) * AN + a] * f;
    out[N_TB + 2 * N_BA + o] = f;
    if (fg) atomicAdd(&cnt_lds, 1);
  }
  __syncthreads();
  if (threadIdx.x == 0 && cnt_lds)
    atomicAdd(&out[N_TB + 3 * N_BA], (float)cnt_lds);
}

// ---------------------------------------------------------------------------
extern "C" void kernel_launch(void* const* d_in, const int* in_sizes, int n_in,
                              void* d_out, int out_size, void* d_ws, size_t ws_size,
                              hipStream_t stream) {
  (void)in_sizes; (void)n_in; (void)out_size; (void)ws_size;
  const float* preds = (const float*)d_in[0];
  const float* gtb   = (const float*)d_in[1];
  const int*   gtc   = (const int*)d_in[2];
  const float* anc   = (const float*)d_in[3];
  float* out = (float*)d_out;

  const size_t nBA = (size_t)BN * AN;
  float*    align_ws  = (float*)d_ws;                       // B*G*A f32 = 68.8 MB
  unsigned* topk_ws   = (unsigned*)(align_ws + (size_t)BN * GN * AN);
  unsigned* inside_ws = topk_ws + nBA;
  int*      bestg_ws  = (int*)(inside_ws + nBA);

  const size_t sum_off = (size_t)BN * AN * 4 + 3 * nBA;

  k1_score_align<<<dim3(AN / 16, BN), 32, 0, stream>>>(
      preds, gtb, gtc, anc, align_ws, topk_ws, inside_ws, bestg_ws,
      out + sum_off);
  k2_topk<<<dim3(GN, BN), 256, 0, stream>>>(align_ws, topk_ws);
  k3_finalize<<<dim3((AN + 255) / 256, BN), 256, 0, stream>>>(
      gtb, gtc, align_ws, topk_ws, inside_ws, bestg_ws, out);
}